// FraudGNN_22548578304372
// MI455X (gfx1250) — compile-verified
//
#include <hip/hip_runtime.h>
#include <math.h>

// ---------------------------------------------------------------------------
// FraudGNN: 2-layer GraphSAGE (2 -> 32 -> 2) + log_softmax, MI455X / gfx1250.
//
// Cost model: dense GEMMs are ~50 MFLOP (negligible); the kernel is bound by
// the layer-2 gather (320 MB) + 80M f32 atomics into a 12.8 MB accumulator
// (L2-resident on the 192 MB L2). Dense transforms use fp32 WMMA
// (v_wmma_f32_16x16x4_f32) so they are exact and effectively free.
// ---------------------------------------------------------------------------

#define N_NODES 100000
#define N_EDGES 2500000
#define N_TILES (N_NODES / 16)   // 6250, exact

typedef __attribute__((ext_vector_type(2))) float v2f;
typedef __attribute__((ext_vector_type(8))) float v8f;

// Workspace layout (floats):
//   deg  : [0,        100000)
//   agg1 : [100000,   300000)
//   agg2 : [300000,   3500000)
//   h    : [3500000,  6700000)
#define WS_DEG   0
#define WS_AGG1  100000
#define WS_AGG2  300000
#define WS_H     3500000
#define WS_ZERO_N 3500000   // deg + agg1 + agg2 must be zeroed each call

// ---------------------------------------------------------------------------
__global__ void __launch_bounds__(256)
zero_ws_kernel(float* __restrict__ ws) {
  int i = blockIdx.x * blockDim.x + threadIdx.x;
  if (i < WS_ZERO_N) ws[i] = 0.0f;
}

// ---------------------------------------------------------------------------
// Edge pass 1: degree + scatter-sum of x[src] (F=2) into agg1[dst].
__global__ void __launch_bounds__(256)
scatter1_kernel(const float* __restrict__ x,
                const int* __restrict__ src,
                const int* __restrict__ dst,
                float* __restrict__ deg,
                float* __restrict__ agg1) {
  int e = blockIdx.x * blockDim.x + threadIdx.x;
  if (e >= N_EDGES) return;
  int s = src[e];
  int d = dst[e];
  atomicAdd(&deg[d], 1.0f);
  atomicAdd(&agg1[2 * d + 0], x[2 * s + 0]);
  atomicAdd(&agg1[2 * d + 1], x[2 * s + 1]);
}

// ---------------------------------------------------------------------------
// Layer 1 dense: h = relu([agg1*dinv | x] @ [W_l1 ; W_r1] + b1), 2->32.
// One wave32 per 16-node tile. A = 16x4 fp32, B = 4x32 (two 4x16 WMMA tiles).
//
// fp32 WMMA A layout (16x4): lanes 0-15 hold M=lane, K=0..1 (VGPR0=K0,VGPR1=K1)
//                            lanes 16-31 hold M=lane-16, K=2..3.
// B (4x16): VGPR0 = row K0 (lanes 0-15) / row K2 (lanes 16-31); VGPR1 = K1/K3.
// C/D (16x16): VGPR r -> M=r (lanes 0-15, N=lane) and M=r+8 (lanes 16-31).
__global__ void __launch_bounds__(256)
sage1_wmma_kernel(const float* __restrict__ x,
                  const float* __restrict__ deg,
                  const float* __restrict__ agg1,
                  const float* __restrict__ Wl1,  // [2,32]
                  const float* __restrict__ Wr1,  // [2,32]
                  const float* __restrict__ b1,   // [32]
                  float* __restrict__ h) {
  int wave = (blockIdx.x * blockDim.x + threadIdx.x) >> 5;  // wave-uniform
  if (wave >= N_TILES) return;                              // uniform exit
  int lane = threadIdx.x & 31;
  int half = lane >> 4;        // 0: K=0,1 (agg side); 1: K=2,3 (root side)
  int m    = lane & 15;
  int base = wave * 16;
  int node = base + m;

  // A fragment: K-concat [agg1 * deg_inv (2) | x (2)]
  v2f a;
  if (half == 0) {
    float dv = 1.0f / fmaxf(deg[node], 1.0f);
    a.x = agg1[2 * node + 0] * dv;
    a.y = agg1[2 * node + 1] * dv;
  } else {
    a.x = x[2 * node + 0];
    a.y = x[2 * node + 1];
  }

  // B rows: half 0 -> W_l1 rows 0..1, half 1 -> W_r1 rows 0..1 (rows 2..3).
  const float* Wb = half ? Wr1 : Wl1;

#pragma unroll
  for (int noff = 0; noff < 32; noff += 16) {
    int col = m + noff;
    v2f b;
    b.x = Wb[col];        // row 0 (or 2) of stacked B
    b.y = Wb[32 + col];   // row 1 (or 3)
    v8f c;
    float cb = b1[col];
#pragma unroll
    for (int r = 0; r < 8; ++r) c[r] = cb;
    c = __builtin_amdgcn_wmma_f32_16x16x4_f32(
        /*neg_a=*/false, a, /*neg_b=*/false, b,
        /*c_mod=*/(short)0, c, /*reuse_a=*/false, /*reuse_b=*/false);
#pragma unroll
    for (int r = 0; r < 8; ++r) {
      int nm = base + r + half * 8;
      h[nm * 32 + col] = fmaxf(c[r], 0.0f);   // relu
    }
  }
}

// ---------------------------------------------------------------------------
// Edge pass 2: scatter-sum of h[src] (F=32) into agg2[dst].
// One wave per edge: lane f handles feature f -> coalesced 128B gather of
// h[src], 32 f32 atomics into the L2-resident 12.8 MB accumulator.
__global__ void __launch_bounds__(256)
scatter2_kernel(const float* __restrict__ h,
                const int* __restrict__ src,
                const int* __restrict__ dst,
                float* __restrict__ agg2) {
  int t = blockIdx.x * blockDim.x + threadIdx.x;   // 80M < 2^31
  if (t >= N_EDGES * 32) return;
  int e = t >> 5;
  int f = t & 31;
  int s = src[e];
  int d = dst[e];
  atomicAdd(&agg2[d * 32 + f], h[s * 32 + f]);
}

// ---------------------------------------------------------------------------
// Layer 2 dense + log_softmax:
//   pre = [agg2*dinv (32) | h (32)] @ [W_l2 ; W_r2]pad + b2   (K=64, N=2 of 16)
//   out = log_softmax(pre, axis=1)
// One wave32 per 16-node tile, 16 chained fp32 WMMAs (K=4 each).
__global__ void __launch_bounds__(256)
sage2_wmma_kernel(const float* __restrict__ h,
                  const float* __restrict__ deg,
                  const float* __restrict__ agg2,
                  const float* __restrict__ Wl2,  // [32,2]
                  const float* __restrict__ Wr2,  // [32,2]
                  const float* __restrict__ b2,   // [2]
                  float* __restrict__ out) {
  int wave = (blockIdx.x * blockDim.x + threadIdx.x) >> 5;  // wave-uniform
  if (wave >= N_TILES) return;
  int lane = threadIdx.x & 31;
  int half = lane >> 4;        // adds +2 to this lane's K pair, +8 to its M row
  int m    = lane & 15;
  int base = wave * 16;
  int node = base + m;
  int n    = m;                // output column this lane covers in B/C/D
  bool nvalid = (n < 2);       // only N=0,1 are real outputs

  float dv = 1.0f / fmaxf(deg[node], 1.0f);
  const float* a2row = agg2 + node * 32;
  const float* hrow  = h + node * 32;

  v8f c;
  float cb = nvalid ? b2[n] : 0.0f;
#pragma unroll
  for (int r = 0; r < 8; ++r) c[r] = cb;

  // K = 0..31 : agg2*dinv  @  W_l2
#pragma unroll
  for (int k = 0; k < 8; ++k) {
    int j = 4 * k + half * 2;                 // this lane's K pair base (<=30)
    v2f a, b;
    a.x = a2row[j] * dv;
    a.y = a2row[j + 1] * dv;
    b.x = nvalid ? Wl2[j * 2 + n] : 0.0f;
    b.y = nvalid ? Wl2[(j + 1) * 2 + n] : 0.0f;
    c = __builtin_amdgcn_wmma_f32_16x16x4_f32(false, a, false, b,
                                              (short)0, c, false, false);
  }
  // K = 32..63 : h  @  W_r2
#pragma unroll
  for (int k = 0; k < 8; ++k) {
    int j = 4 * k + half * 2;
    v2f a, b;
    a.x = hrow[j];
    a.y = hrow[j + 1];
    b.x = nvalid ? Wr2[j * 2 + n] : 0.0f;
    b.y = nvalid ? Wr2[(j + 1) * 2 + n] : 0.0f;
    c = __builtin_amdgcn_wmma_f32_16x16x4_f32(false, a, false, b,
                                              (short)0, c, false, false);
  }

  // log_softmax over the 2 classes. In the D layout, N=0 and N=1 for a given
  // row M live in adjacent lanes (0,1) and (16,17) -> pair-swap via shfl_xor.
#pragma unroll
  for (int r = 0; r < 8; ++r) {
    float v = c[r];
    float p = __shfl_xor(v, 1, 32);           // partner class value
    if (nvalid) {
      float mx  = fmaxf(v, p);
      float lse = mx + logf(__expf(v - mx) + __expf(p - mx));
      int nm = base + r + half * 8;
      out[nm * 2 + n] = v - lse;
    }
  }
}

// ---------------------------------------------------------------------------
extern "C" void kernel_launch(void* const* d_in, const int* in_sizes, int n_in,
                              void* d_out, int out_size, void* d_ws, size_t ws_size,
                              hipStream_t stream) {
  const float* x   = (const float*)d_in[0];
  const int*   ei  = (const int*)d_in[1];   // [2, N_EDGES] row-major
  const float* Wl1 = (const float*)d_in[2];
  const float* Wr1 = (const float*)d_in[3];
  const float* b1  = (const float*)d_in[4];
  const float* Wl2 = (const float*)d_in[5];
  const float* Wr2 = (const float*)d_in[6];
  const float* b2  = (const float*)d_in[7];
  float* out = (float*)d_out;

  float* ws   = (float*)d_ws;
  float* deg  = ws + WS_DEG;
  float* agg1 = ws + WS_AGG1;
  float* agg2 = ws + WS_AGG2;
  float* h    = ws + WS_H;
  const int* src = ei;
  const int* dst = ei + N_EDGES;

  // 1) zero accumulators (deg, agg1, agg2)
  zero_ws_kernel<<<(WS_ZERO_N + 255) / 256, 256, 0, stream>>>(ws);

  // 2) degree + layer-1 scatter (F=2)
  scatter1_kernel<<<(N_EDGES + 255) / 256, 256, 0, stream>>>(x, src, dst, deg, agg1);

  // 3) layer-1 dense via fp32 WMMA (8 waves/block -> 8 tiles/block)
  int tile_blocks = (N_TILES + 7) / 8;   // 782
  sage1_wmma_kernel<<<tile_blocks, 256, 0, stream>>>(x, deg, agg1, Wl1, Wr1, b1, h);

  // 4) layer-2 scatter (F=32, one lane per feature)
  int s2_threads = N_EDGES * 32;         // 80M
  scatter2_kernel<<<s2_threads / 256, 256, 0, stream>>>(h, src, dst, agg2);

  // 5) layer-2 dense + log_softmax via fp32 WMMA
  sage2_wmma_kernel<<<tile_blocks, 256, 0, stream>>>(h, deg, agg2, Wl2, Wr2, b2, out);
}